// HyperCubeMoE_73684458930970
// MI455X (gfx1250) — compile-verified
//
#include <hip/hip_runtime.h>
#include <hip/hip_bf16.h>

// ---------------------------------------------------------------------------
// HyperCube MoE, f32 end-to-end on gfx1250 (MI455X).
// Memory-bound (~320 MB traffic vs ~1.1 TFLOP of f32 WMMA work), so we keep
// exact f32 reference numerics and drive V_WMMA_F32_16X16X4_F32.
// ---------------------------------------------------------------------------

typedef __attribute__((ext_vector_type(2))) float v2f;
typedef __attribute__((ext_vector_type(8))) float v8f;

#define DIN   1024
#define DOUT  1024
#define HDIM  4096
#define NEXP  8
#define MT    32      // rows per workgroup (2 WMMA row tiles)
#define KCH   128     // H-chunk (g buffer width); 16 h-cols per wave in GEMM1

static __device__ __forceinline__ v8f wmma_f32(v2f a, v2f b, v8f c) {
    // 8 args: (neg_a, A, neg_b, B, c_mod, C, reuse_a, reuse_b)
    return __builtin_amdgcn_wmma_f32_16x16x4_f32(false, a, false, b,
                                                 (short)0, c, false, false);
}

// ---------------------------------------------------------------------------
// Gating: one wave32 per token. logits = relu(x@Wg); scores = logits @ mask^T
// (mask = hypercube corners, sign = bit g of expert id); top-2 of softmax
// renormalized == exp(sa)/(exp(sa)+exp(sb)). Writes dense combine [N, 8].
// ---------------------------------------------------------------------------
__global__ __launch_bounds__(256) void hcmoe_gate_kernel(
    const float* __restrict__ x, const float* __restrict__ Wg,
    float* __restrict__ combine, int N) {
    const int wave = threadIdx.x >> 5;
    const int lane = threadIdx.x & 31;
    const int row  = blockIdx.x * 8 + wave;
    if (row >= N) return;

    const float* xr = x + (size_t)row * DIN;
    float a0 = 0.f, a1 = 0.f, a2 = 0.f;
    for (int k = lane; k < DIN; k += 32) {
        const float xv = xr[k];
        a0 = fmaf(xv, Wg[k * 3 + 0], a0);
        a1 = fmaf(xv, Wg[k * 3 + 1], a1);
        a2 = fmaf(xv, Wg[k * 3 + 2], a2);
    }
    #pragma unroll
    for (int off = 16; off > 0; off >>= 1) {   // wave32 butterfly reduction
        a0 += __shfl_xor(a0, off, 32);
        a1 += __shfl_xor(a1, off, 32);
        a2 += __shfl_xor(a2, off, 32);
    }
    if (lane == 0) {
        const float l0 = fmaxf(a0, 0.f), l1 = fmaxf(a1, 0.f), l2 = fmaxf(a2, 0.f);
        float s[NEXP];
        #pragma unroll
        for (int c = 0; c < NEXP; ++c) {
            s[c] = ((c & 1) ? l0 : -l0) + ((c & 2) ? l1 : -l1) + ((c & 4) ? l2 : -l2);
        }
        int i1 = 0;
        #pragma unroll
        for (int c = 1; c < NEXP; ++c) if (s[c] > s[i1]) i1 = c;
        int i2 = (i1 == 0) ? 1 : 0;
        #pragma unroll
        for (int c = 0; c < NEXP; ++c) if (c != i1 && s[c] > s[i2]) i2 = c;
        const float e2 = __expf(s[i2] - s[i1]);      // e1 == 1
        const float inv = 1.f / (1.f + e2);
        float* cr = combine + (size_t)row * NEXP;
        #pragma unroll
        for (int c = 0; c < NEXP; ++c) cr[c] = 0.f;
        cr[i1] = inv;
        cr[i2] = e2 * inv;
    }
}

// ---------------------------------------------------------------------------
// Fused experts: 256 threads = 8 waves, 32-row tile per workgroup.
// x tile lives in LDS for the whole kernel; per (expert, H-chunk):
//   GEMM1 (wave -> 32x16 h slice, K=1024) -> relu/b1/combine-scale -> LDS g
//   GEMM2 (wave -> 32x128 out slice accum, K=128)
// ---------------------------------------------------------------------------
__global__ __launch_bounds__(256) void hcmoe_expert_kernel(
    const float* __restrict__ x,
    const float* __restrict__ W1, const float* __restrict__ b1,
    const float* __restrict__ W2, const float* __restrict__ b2,
    const float* __restrict__ combine, float* __restrict__ out) {

    __shared__ float xs[MT][DIN];    // 128 KB: x tile, loaded once
    __shared__ float gs[MT][KCH];    // 16 KB : combine-scaled relu(h) chunk
    __shared__ float cs[MT][NEXP];   // 1 KB  : combine weights for the tile

    const int tid  = threadIdx.x;
    const int wave = tid >> 5;       // 0..7
    const int lane = tid & 31;
    const int l16  = lane & 15;      // N (or M) index within a 16-wide tile
    const int hi   = lane >> 4;      // lane-half: selects K pair / M+8 rows
    const int r0   = blockIdx.x * MT;

    // --- stage x tile (32 x 1024 f32) and combine tile into LDS ---
    for (int i = tid; i < MT * DIN / 4; i += 256) {
        ((float4*)&xs[0][0])[i] = ((const float4*)x)[(size_t)r0 * (DIN / 4) + i];
    }
    if (tid < MT * NEXP) {
        cs[tid >> 3][tid & 7] = combine[(size_t)(r0 + (tid >> 3)) * NEXP + (tid & 7)];
    }
    __syncthreads();

    // persistent output accumulators: rows [0..32), cols [wave*128, wave*128+128)
    v8f oacc[2][8];
    #pragma unroll
    for (int rt = 0; rt < 2; ++rt)
        #pragma unroll
        for (int ct = 0; ct < 8; ++ct) oacc[rt][ct] = (v8f){};

    for (int e = 0; e < NEXP; ++e) {
        const float* __restrict__ W1e = W1 + (size_t)e * DIN * HDIM;
        const float* __restrict__ W2e = W2 + (size_t)e * HDIM * DOUT;
        const float* __restrict__ b1e = b1 + (size_t)e * HDIM;
        const float* __restrict__ b2e = b2 + (size_t)e * DOUT;

        // combine weights for the rows this lane's C-layout elements map to:
        // C/D layout: VGPR i, lane-half hi -> row = rt*16 + i + 8*hi
        float cvals[2][8];
        #pragma unroll
        for (int rt = 0; rt < 2; ++rt)
            #pragma unroll
            for (int i = 0; i < 8; ++i) cvals[rt][i] = cs[rt * 16 + i + 8 * hi][e];

        for (int hc = 0; hc < HDIM; hc += KCH) {
            // ---------------- GEMM1: h[:, hc + wave*16 + l16] ----------------
            const int hcol = hc + wave * 16 + l16;
            v8f h0 = (v8f){}, h1 = (v8f){};
            // B layout: VGPR0/1 hold rows K=(k+2*hi), K=(k+2*hi+1) at col=l16
            const float* bp = W1e + (size_t)(2 * hi) * HDIM + hcol;
            #pragma unroll 4
            for (int k = 0; k < DIN; k += 4) {
                v2f b;
                b.x = bp[(size_t)k * HDIM];
                b.y = bp[(size_t)(k + 1) * HDIM];
                // A layout: row = l16, K pair (k+2*hi, k+2*hi+1), 8B aligned
                const v2f a0 = *(const v2f*)&xs[l16][k + 2 * hi];
                const v2f a1 = *(const v2f*)&xs[16 + l16][k + 2 * hi];
                h0 = wmma_f32(a0, b, h0);
                h1 = wmma_f32(a1, b, h1);
            }

            __syncthreads();   // previous chunk's GEMM2 reads of gs are done
            const float bias1 = b1e[hcol];
            #pragma unroll
            for (int i = 0; i < 8; ++i) {
                float v0 = fmaxf(h0[i] + bias1, 0.f);
                float v1 = fmaxf(h1[i] + bias1, 0.f);
                gs[i + 8 * hi][wave * 16 + l16]      = v0 * cvals[0][i];
                gs[16 + i + 8 * hi][wave * 16 + l16] = v1 * cvals[1][i];
            }
            __syncthreads();

            // ---------------- GEMM2: oacc += g[32,128] @ W2e[hc:hc+128, :] ---
            const int ocol0 = wave * 128 + l16;
            const float* b2p = W2e + (size_t)(hc + 2 * hi) * DOUT + ocol0;
            #pragma unroll 2
            for (int k = 0; k < KCH; k += 4) {
                const v2f a0 = *(const v2f*)&gs[l16][k + 2 * hi];
                const v2f a1 = *(const v2f*)&gs[16 + l16][k + 2 * hi];
                const float* bk = b2p + (size_t)k * DOUT;
                #pragma unroll
                for (int ct = 0; ct < 8; ++ct) {
                    v2f b;
                    b.x = bk[ct * 16];
                    b.y = bk[ct * 16 + DOUT];
                    oacc[0][ct] = wmma_f32(a0, b, oacc[0][ct]);
                    oacc[1][ct] = wmma_f32(a1, b, oacc[1][ct]);
                }
            }
        }

        // bias2 term: out += c[row] * b2e[col]
        #pragma unroll
        for (int ct = 0; ct < 8; ++ct) {
            const float bb = b2e[wave * 128 + ct * 16 + l16];
            #pragma unroll
            for (int i = 0; i < 8; ++i) {
                oacc[0][ct][i] = fmaf(cvals[0][i], bb, oacc[0][ct][i]);
                oacc[1][ct][i] = fmaf(cvals[1][i], bb, oacc[1][ct][i]);
            }
        }
    }

    // --- write back: C layout -> out[r0+row][wave*128 + ct*16 + l16] ---
    #pragma unroll
    for (int rt = 0; rt < 2; ++rt)
        #pragma unroll
        for (int ct = 0; ct < 8; ++ct)
            #pragma unroll
            for (int i = 0; i < 8; ++i) {
                const int row = r0 + rt * 16 + i + 8 * hi;
                const int col = wave * 128 + ct * 16 + l16;
                out[(size_t)row * DOUT + col] = oacc[rt][ct][i];
            }
}

extern "C" void kernel_launch(void* const* d_in, const int* in_sizes, int n_in,
                              void* d_out, int out_size, void* d_ws, size_t ws_size,
                              hipStream_t stream) {
    const float* x  = (const float*)d_in[0];
    const float* Wg = (const float*)d_in[1];
    const float* W1 = (const float*)d_in[2];
    const float* b1 = (const float*)d_in[3];
    const float* W2 = (const float*)d_in[4];
    const float* b2 = (const float*)d_in[5];
    float* out = (float*)d_out;

    const int N = in_sizes[0] / DIN;          // 8192
    float* combine = (float*)d_ws;            // N * 8 f32 = 256 KB scratch

    hcmoe_gate_kernel<<<(N + 7) / 8, 256, 0, stream>>>(x, Wg, combine, N);
    hcmoe_expert_kernel<<<N / MT, 256, 0, stream>>>(x, W1, b1, W2, b2, combine, out);
}